// cola_gnn_68255620268677
// MI455X (gfx1250) — compile-verified
//
#include <hip/hip_runtime.h>
#include <math.h>

// MI455X / gfx1250, wave32. f32 WMMA path (V_WMMA_F32_16X16X4_F32).
typedef float v2f __attribute__((ext_vector_type(2)));
typedef float v8f __attribute__((ext_vector_type(8)));

// Lane layouts per CDNA5 ISA 7.12.2 (f32, 16x16x4):
//   A 16x4:  lanes 0-15 -> M=lane, VGPR{0,1}=K{0,1}; lanes 16-31 -> K{2,3}
//   B 4x16:  VGPR v: lanes 0-15 K=v, lanes 16-31 K=v+2 ; N = lane&15
//   C/D:     VGPR v: lanes 0-15 M=v, lanes 16-31 M=v+8 ; N = lane&15
#define WMMA_F32(acc, a, b) \
    __builtin_amdgcn_wmma_f32_16x16x4_f32(false, (a), false, (b), (short)0, (acc), false, false)

__device__ __forceinline__ v2f load_afrag(const float* __restrict__ arow, int k0, int koff) {
    v2f a; a.x = arow[k0 + koff]; a.y = arow[k0 + koff + 1]; return a;
}
__device__ __forceinline__ v2f load_bfrag(const float* __restrict__ bcol, int k0, int koff, int ldb) {
    v2f b; b.x = bcol[(k0 + koff) * ldb]; b.y = bcol[(k0 + koff + 1) * ldb]; return b;
}

// Single 16x16 tile (used for the small-N spatial GEMM)
__device__ __forceinline__ v8f wmma_tile_f32(const float* __restrict__ Abase, int lda,
                                             const float* __restrict__ Bbase, int ldb,
                                             int Klen) {
    const int lane = threadIdx.x & 31;
    const int koff = (lane >> 4) << 1;
    const int l15  = lane & 15;
    const float* arow = Abase + l15 * lda;
    const float* bcol = Bbase + l15;
    v8f acc = {0.f, 0.f, 0.f, 0.f, 0.f, 0.f, 0.f, 0.f};
    for (int k0 = 0; k0 < Klen; k0 += 4)
        acc = WMMA_F32(acc, load_afrag(arow, k0, koff), load_bfrag(bcol, k0, koff, ldb));
    return acc;
}

// --- 1. RNN over 2048 sequences (one 64-thread block per sequence) ---------
__global__ void rnn_kernel(const float* __restrict__ x, const float* __restrict__ W_ih,
                           const float* __restrict__ W_hh, const float* __restrict__ b_ih,
                           const float* __restrict__ b_hh, float* __restrict__ last_hid) {
    const int n = blockIdx.x;          // n = b*512 + m
    const int j = threadIdx.x;         // 0..63
    const int b = n >> 9, m = n & 511;
    __shared__ float Wtr[64 * 64];     // Wtr[k*64+j] = W_hh[j][k]
    __shared__ float hs[64];
    for (int k = 0; k < 64; ++k) Wtr[k * 64 + j] = W_hh[j * 64 + k];
    hs[j] = 0.f;
    const float wih = W_ih[j], bih = b_ih[j], bhh = b_hh[j];
    __syncthreads();
    for (int t = 0; t < 20; ++t) {
        const float xv = x[b * (20 * 512) + t * 512 + m];
        float acc = fmaf(xv, wih, bih) + bhh;
#pragma unroll
        for (int k = 0; k < 64; ++k) acc = fmaf(Wtr[k * 64 + j], hs[k], acc);
        __syncthreads();
        hs[j] = tanhf(acc);
        __syncthreads();
    }
    last_hid[n * 64 + j] = hs[j];
}

// --- 2. p = lh @ W1^T, q = lh @ W2^T  (HALF=32 each) -----------------------
__global__ void pq_kernel(const float* __restrict__ lh, const float* __restrict__ W1,
                          const float* __restrict__ W2, float* __restrict__ p,
                          float* __restrict__ q) {
    const int n = blockIdx.x, tid = threadIdx.x;   // block 64
    __shared__ float h[64];
    h[tid] = lh[n * 64 + tid];
    __syncthreads();
    const float* Wrow = (tid < 32) ? (W1 + tid * 64) : (W2 + (tid - 32) * 64);
    float s = 0.f;
#pragma unroll
    for (int k = 0; k < 64; ++k) s = fmaf(Wrow[k], h[k], s);
    if (tid < 32) p[n * 32 + tid] = s;
    else          q[n * 32 + (tid - 32)] = s;
}

// --- 3. a_mx[b,i,j] = V . elu(p[b,i,:] + q[b,j,:] + b1) + bv ---------------
__global__ void attn_kernel(const float* __restrict__ p, const float* __restrict__ q,
                            const float* __restrict__ b1, const float* __restrict__ V,
                            const float* __restrict__ bv, float* __restrict__ amx) {
    const int idx = blockIdx.x * blockDim.x + threadIdx.x;   // 4*512*512
    const int b = idx >> 18;
    const int i = (idx >> 9) & 511;
    const int j = idx & 511;
    const float* pr = p + (b * 512 + i) * 32;
    const float* qr = q + (b * 512 + j) * 32;
    float s = 0.f;
#pragma unroll
    for (int h = 0; h < 32; ++h) {
        float e = pr[h] + qr[h] + b1[h];
        e = (e > 0.f) ? e : (expf(e) - 1.f);     // elu, alpha=1
        s = fmaf(e, V[h], s);
    }
    amx[idx] = s + bv[0];
}

// --- 4/5. column L2 norms (axis=1) and in-place normalize ------------------
__global__ void colnorm_kernel(const float* __restrict__ amx, float* __restrict__ nrm) {
    const int idx = blockIdx.x * blockDim.x + threadIdx.x;   // 2048 = (b, j)
    const int b = idx >> 9, j = idx & 511;
    const float* col = amx + b * 512 * 512 + j;
    float s = 0.f;
    for (int i = 0; i < 512; ++i) { const float v = col[i * 512]; s = fmaf(v, v, s); }
    nrm[idx] = fmaxf(sqrtf(s), 1e-12f);
}

__global__ void scale_kernel(const float* __restrict__ nrm, float* __restrict__ amx) {
    const int idx = blockIdx.x * blockDim.x + threadIdx.x;
    const int b = idx >> 18, j = idx & 511;
    amx[idx] /= nrm[b * 512 + j];
}

// --- 6. WMMA GEMM (2x2 register-blocked, 32x32 tile/wave):
//        T = a_mx[b] @ Wb ; A = adj*sig(T+wb) + a_mx*(1-sig) ----------------
__device__ __forceinline__ void gate_epilogue(v8f acc, const float* __restrict__ Ab,
                                              const float* __restrict__ adj,
                                              float* __restrict__ Aout,
                                              int ib, int jb, float wbs,
                                              int half, int l15) {
#pragma unroll
    for (int v = 0; v < 8; ++v) {
        const int i = ib + v + half * 8;
        const int j = jb + l15;
        const float c = 1.f / (1.f + expf(-(acc[v] + wbs)));
        const float av = Ab[i * 512 + j];
        Aout[i * 512 + j] = fmaf(adj[i * 512 + j], c, av * (1.f - c));
    }
}

__global__ void gate_gemm_kernel(const float* __restrict__ amx, const float* __restrict__ Wb,
                                 const float* __restrict__ wb, const float* __restrict__ adj,
                                 float* __restrict__ A) {
    const int b  = blockIdx.z;
    const int i0 = blockIdx.y * 32;
    const int j0 = blockIdx.x * 32;
    const float* Ab = amx + b * 512 * 512;
    const int lane = threadIdx.x & 31;
    const int half = lane >> 4, l15 = lane & 15;
    const int koff = half << 1;
    const float* arow0 = Ab + (i0 + l15) * 512;
    const float* arow1 = arow0 + 16 * 512;
    const float* bcol0 = Wb + j0 + l15;
    const float* bcol1 = bcol0 + 16;
    v8f acc00 = {0.f,0.f,0.f,0.f,0.f,0.f,0.f,0.f}, acc01 = acc00;
    v8f acc10 = acc00, acc11 = acc00;
    for (int k0 = 0; k0 < 512; k0 += 4) {
        const v2f a0 = load_afrag(arow0, k0, koff);
        const v2f a1 = load_afrag(arow1, k0, koff);
        const v2f b0 = load_bfrag(bcol0, k0, koff, 512);
        const v2f b1 = load_bfrag(bcol1, k0, koff, 512);
        acc00 = WMMA_F32(acc00, a0, b0);
        acc01 = WMMA_F32(acc01, a0, b1);
        acc10 = WMMA_F32(acc10, a1, b0);
        acc11 = WMMA_F32(acc11, a1, b1);
    }
    float* Aout = A + b * 512 * 512;
    const float wbs = wb[0];
    gate_epilogue(acc00, Ab, adj, Aout, i0,      j0,      wbs, half, l15);
    gate_epilogue(acc01, Ab, adj, Aout, i0,      j0 + 16, wbs, half, l15);
    gate_epilogue(acc10, Ab, adj, Aout, i0 + 16, j0,      wbs, half, l15);
    gate_epilogue(acc11, Ab, adj, Aout, i0 + 16, j0 + 16, wbs, half, l15);
}

// --- 7. convs (K=10; dil-2 K=10 -> 2 outs), relu, then r_l @ gc1_w ---------
__global__ void conv_h1_kernel(const float* __restrict__ x, const float* __restrict__ conv_w,
                               const float* __restrict__ conv_b, const float* __restrict__ convl_w,
                               const float* __restrict__ convl_b, const float* __restrict__ gc1_w,
                               float* __restrict__ h1) {
    const int n = blockIdx.x, tid = threadIdx.x;   // block 64
    const int b = n >> 9, m = n & 511;
    __shared__ float xr[20];
    __shared__ float rl[30];
    if (tid < 20) xr[tid] = x[b * 10240 + tid * 512 + m];
    __syncthreads();
    if (tid < 10) {                       // full-width conv, out len 1
        float s = conv_b[tid];
#pragma unroll
        for (int t = 0; t < 20; ++t) s = fmaf(xr[t], conv_w[tid * 20 + t], s);
        rl[tid * 3] = fmaxf(s, 0.f);
    } else if (tid < 30) {                // dilation-2 conv, out len 2
        const int k = (tid - 10) >> 1, pos = (tid - 10) & 1;
        float s = convl_b[k];
#pragma unroll
        for (int t = 0; t < 10; ++t) s = fmaf(xr[pos + 2 * t], convl_w[k * 10 + t], s);
        rl[k * 3 + 1 + pos] = fmaxf(s, 0.f);
    }
    __syncthreads();
    float s = 0.f;
#pragma unroll
    for (int f = 0; f < 30; ++f) s = fmaf(rl[f], gc1_w[f * 64 + tid], s);
    h1[n * 64 + tid] = s;
}

// --- 8. WMMA GEMM (1x4 blocked, 16x64 tile/wave): xg = relu(A@h1 + gc1_b) --
__global__ void xg_gemm_kernel(const float* __restrict__ A, const float* __restrict__ h1,
                               const float* __restrict__ gc1_b, float* __restrict__ xg) {
    const int b  = blockIdx.z;
    const int i0 = blockIdx.y * 16;
    const int lane = threadIdx.x & 31;
    const int half = lane >> 4, l15 = lane & 15;
    const int koff = half << 1;
    const float* arow = A + b * 512 * 512 + (i0 + l15) * 512;
    const float* Bb   = h1 + b * 512 * 64 + l15;     // ldb = 64
    v8f acc0 = {0.f,0.f,0.f,0.f,0.f,0.f,0.f,0.f};
    v8f acc1 = acc0, acc2 = acc0, acc3 = acc0;
    for (int k0 = 0; k0 < 512; k0 += 4) {
        const v2f a  = load_afrag(arow, k0, koff);
        acc0 = WMMA_F32(acc0, a, load_bfrag(Bb,      k0, koff, 64));
        acc1 = WMMA_F32(acc1, a, load_bfrag(Bb + 16, k0, koff, 64));
        acc2 = WMMA_F32(acc2, a, load_bfrag(Bb + 32, k0, koff, 64));
        acc3 = WMMA_F32(acc3, a, load_bfrag(Bb + 48, k0, koff, 64));
    }
    v8f accs[4] = {acc0, acc1, acc2, acc3};
#pragma unroll
    for (int t = 0; t < 4; ++t) {
        const int j = t * 16 + l15;
        const float bj = gc1_b[j];
#pragma unroll
        for (int v = 0; v < 8; ++v) {
            const int i = i0 + v + half * 8;
            xg[(b * 512 + i) * 64 + j] = fmaxf(accs[t][v] + bj, 0.f);
        }
    }
}

// --- 9. sp2 = xg @ gc2_w, padded to N=16 (cols 10..15 zero) ----------------
__global__ void sp2_kernel(const float* __restrict__ xg, const float* __restrict__ gc2_w,
                           float* __restrict__ sp2) {
    const int idx = blockIdx.x * blockDim.x + threadIdx.x;   // 2048*16
    const int n = idx >> 4, j = idx & 15;
    float s = 0.f;
    if (j < 10) {
        const float* xr = xg + n * 64;
#pragma unroll
        for (int k = 0; k < 64; ++k) s = fmaf(xr[k], gc2_w[k * 10 + j], s);
    }
    sp2[idx] = s;
}

// --- 10. WMMA GEMM: out_spatial = relu(A[b] @ sp2[b] + gc2_b), N=10 --------
__global__ void spat_gemm_kernel(const float* __restrict__ A, const float* __restrict__ sp2,
                                 const float* __restrict__ gc2_b, float* __restrict__ outsp) {
    const int b  = blockIdx.z;
    const int i0 = blockIdx.y * 16;
    v8f acc = wmma_tile_f32(A + b * 512 * 512 + i0 * 512, 512, sp2 + b * 512 * 16, 16, 512);
    const int lane = threadIdx.x & 31;
    const int half = lane >> 4, l15 = lane & 15;
    if (l15 < 10) {
#pragma unroll
        for (int v = 0; v < 8; ++v) {
            const int i = i0 + v + half * 8;
            outsp[(b * 512 + i) * 10 + l15] = fmaxf(acc[v] + gc2_b[l15], 0.f);
        }
    }
}

// --- 11. out[n] = [out_spatial | last_hid] . out_w + out_b -----------------
__global__ void final_kernel(const float* __restrict__ outsp, const float* __restrict__ lh,
                             const float* __restrict__ out_w, const float* __restrict__ out_b,
                             float* __restrict__ out) {
    const int n = blockIdx.x * blockDim.x + threadIdx.x;   // 2048
    float s = out_b[0];
#pragma unroll
    for (int j = 0; j < 10; ++j) s = fmaf(outsp[n * 10 + j], out_w[j], s);
#pragma unroll
    for (int h = 0; h < 64; ++h) s = fmaf(lh[n * 64 + h], out_w[10 + h], s);
    out[n] = s;
}

extern "C" void kernel_launch(void* const* d_in, const int* in_sizes, int n_in,
                              void* d_out, int out_size, void* d_ws, size_t ws_size,
                              hipStream_t stream) {
    const float* x       = (const float*)d_in[0];
    const float* W_ih    = (const float*)d_in[1];
    const float* W_hh    = (const float*)d_in[2];
    const float* b_ih    = (const float*)d_in[3];
    const float* b_hh    = (const float*)d_in[4];
    const float* W1      = (const float*)d_in[5];
    const float* b1      = (const float*)d_in[6];
    const float* W2      = (const float*)d_in[7];
    const float* V       = (const float*)d_in[8];
    const float* bv      = (const float*)d_in[9];
    const float* Wb      = (const float*)d_in[10];
    const float* wb      = (const float*)d_in[11];
    const float* conv_w  = (const float*)d_in[12];
    const float* conv_b  = (const float*)d_in[13];
    const float* convl_w = (const float*)d_in[14];
    const float* convl_b = (const float*)d_in[15];
    const float* gc1_w   = (const float*)d_in[16];
    const float* gc1_b   = (const float*)d_in[17];
    const float* gc2_w   = (const float*)d_in[18];
    const float* gc2_b   = (const float*)d_in[19];
    const float* out_w   = (const float*)d_in[20];
    const float* out_b   = (const float*)d_in[21];
    const float* adj     = (const float*)d_in[22];

    float* ws = (float*)d_ws;                       // ~10.7 MB total, L2-resident
    float* last_hid = ws; ws += 2048 * 64;
    float* p        = ws; ws += 2048 * 32;
    float* q        = ws; ws += 2048 * 32;
    float* amx      = ws; ws += 4 * 512 * 512;
    float* nrm      = ws; ws += 2048;
    float* Amat     = ws; ws += 4 * 512 * 512;
    float* h1       = ws; ws += 2048 * 64;
    float* xg       = ws; ws += 2048 * 64;
    float* sp2      = ws; ws += 2048 * 16;
    float* outsp    = ws; ws += 2048 * 10;

    rnn_kernel      <<<2048, 64, 0, stream>>>(x, W_ih, W_hh, b_ih, b_hh, last_hid);
    pq_kernel       <<<2048, 64, 0, stream>>>(last_hid, W1, W2, p, q);
    attn_kernel     <<<4096, 256, 0, stream>>>(p, q, b1, V, bv, amx);
    colnorm_kernel  <<<8, 256, 0, stream>>>(amx, nrm);
    scale_kernel    <<<4096, 256, 0, stream>>>(nrm, amx);
    gate_gemm_kernel<<<dim3(16, 16, 4), 32, 0, stream>>>(amx, Wb, wb, adj, Amat);
    conv_h1_kernel  <<<2048, 64, 0, stream>>>(x, conv_w, conv_b, convl_w, convl_b, gc1_w, h1);
    xg_gemm_kernel  <<<dim3(1, 32, 4), 32, 0, stream>>>(Amat, h1, gc1_b, xg);
    sp2_kernel      <<<128, 256, 0, stream>>>(xg, gc2_w, sp2);
    spat_gemm_kernel<<<dim3(1, 32, 4), 32, 0, stream>>>(Amat, sp2, gc2_b, outsp);
    final_kernel    <<<8, 256, 0, stream>>>(outsp, last_hid, out_w, out_b, (float*)d_out);
}